// QPSolver_50465865728223
// MI455X (gfx1250) — compile-verified
//
#include <hip/hip_runtime.h>

// QP fixed-point solver on gfx1250 (MI455X).
// Math: tD = (I + H P^-1 H^T)^-1; per step (alpha=beta=1):
//   s = x1 + x2; u = tD s; x1' = u + mu; x2' = relu(s - 2 x1'); p = pinvH x2' - pinvH b
// One workgroup per batch (bs=256). Precompute GEMMs + per-iteration matvecs use
// V_WMMA_F32_16X16X4_F32. SPD inverses via in-place Gauss-Jordan sweep (f32).
// The loop-invariant matrices tD / pinvH are preloaded into VGPRs so the 1000
// sequential iterations issue WMMA from registers with only vector LDS reads.

#define NDIM 64
#define MDIM 128
#define TWOM 256
#define BLK  256

typedef float v2f __attribute__((ext_vector_type(2)));
typedef float v8f __attribute__((ext_vector_type(8)));

__device__ inline v8f wmma4(v2f a, v2f b, v8f c) {
  // D(16x16,f32) = A(16x4,f32) * B(4x16,f32) + C
  return __builtin_amdgcn_wmma_f32_16x16x4_f32(false, a, false, b, (short)0, c,
                                               false, false);
}

// 16x16 tile of C = A(MxK) * B(KxN), generic strides (transposes for free).
__device__ inline void gemm_tile(const float* __restrict__ A, int sAr, int sAc,
                                 const float* __restrict__ B, int sBr, int sBc,
                                 float* __restrict__ C, int ldc,
                                 int m0, int n0, int K, int lane) {
  const int mr = lane & 15;
  const int h  = lane >> 4;
  v8f c = {};
  for (int k0 = 0; k0 < K; k0 += 4) {
    v2f a, b;
    a.x = A[(m0 + mr) * sAr + (k0 + 2 * h)     * sAc];
    a.y = A[(m0 + mr) * sAr + (k0 + 2 * h + 1) * sAc];
    b.x = B[(k0 + 2 * h)     * sBr + (n0 + mr) * sBc];
    b.y = B[(k0 + 2 * h + 1) * sBr + (n0 + mr) * sBc];
    c = wmma4(a, b, c);
  }
#pragma unroll
  for (int r = 0; r < 8; ++r)
    C[(m0 + r + 8 * h) * ldc + n0 + mr] = c[r];
}

// In-place inverse of SPD n x n matrix via Gauss-Jordan sweep operator.
// After sweeping all pivots A -> -A^{-1}; final pass negates.
__device__ inline void sweep_inverse(float* A, int n, int log2n,
                                     float* rowk, float* colk, int tid) {
  const int nn = n * n;
  for (int k = 0; k < n; ++k) {
    for (int i = tid; i < n; i += BLK) {
      rowk[i] = A[k * n + i];
      colk[i] = A[i * n + k];
    }
    __syncthreads();
    const float invd = 1.0f / rowk[k];
    for (int e = tid; e < nn; e += BLK) {
      const int i = e >> log2n;
      const int j = e & (n - 1);
      float v;
      if (i == k)      v = (j == k) ? -invd : rowk[j] * invd;
      else if (j == k) v = colk[i] * invd;
      else             v = A[e] - colk[i] * rowk[j] * invd;
      A[e] = v;
    }
    __syncthreads();
  }
  for (int e = tid; e < nn; e += BLK) A[e] = -A[e];
  __syncthreads();
}

__global__ void QPSolver_kernel(const float* __restrict__ q_g,
                                const float* __restrict__ b_g,
                                const float* __restrict__ P_g,
                                const float* __restrict__ H_g,
                                const int*   __restrict__ iters_p,
                                float* __restrict__ out, int bs) {
  extern __shared__ float smem[];
  float* sTD  = smem;                    // 128x128: S -> I+S -> tD
  float* sH   = sTD  + MDIM * MDIM;      // 128x64 : H (row major)
  float* sPHt = sH   + MDIM * NDIM;      // 64x128 : PinvHt, later pinvH
  float* sP   = sPHt + NDIM * MDIM;      // 64x64  : P->Pinv, later HtH->Ginv
  float* qv   = sP + NDIM * NDIM;        // 64
  float* bv   = qv + 64;                 // 128
  float* pq   = bv + 128;                // 64
  float* hq   = pq + 64;                 // 128
  float* mu   = hq + 128;                // 128
  float* hb   = mu + 128;                // 64
  float* x1   = hb + 64;                 // 128
  float* x2   = x1 + 128;                // 128
  float* sv   = x2 + 128;                // 128
  float* uv   = sv + 128;                // 128
  float* rowk = uv + 128;                // 128 (GJ scratch; pv partials in loop)
  float* colk = rowk + 128;              // 128

  const int tid  = threadIdx.x;
  const int lane = tid & 31;
  const int wave = __builtin_amdgcn_readfirstlane(tid >> 5);  // wave-uniform SGPR
  const int mr   = lane & 15;
  const int h    = lane >> 4;
  const int bid  = blockIdx.x;
  const int iters = iters_p[0];

  // ---- stage inputs into LDS (prefetch the bulk first) ----
  const float* Pg = P_g + (size_t)bid * NDIM * NDIM;
  const float* Hg = H_g + (size_t)bid * MDIM * NDIM;
  __builtin_prefetch(&Pg[tid * 16], 0, 3);   // global_prefetch_b8
  __builtin_prefetch(&Hg[tid * 32], 0, 3);
  for (int e = tid; e < NDIM * NDIM; e += BLK) sP[e] = Pg[e];
  for (int e = tid; e < MDIM * NDIM; e += BLK) sH[e] = Hg[e];
  if (tid < NDIM) qv[tid] = q_g[(size_t)bid * NDIM + tid];
  if (tid < MDIM) bv[tid] = b_g[(size_t)bid * MDIM + tid];
  __syncthreads();

  // ---- Pinv = P^{-1} ----
  sweep_inverse(sP, NDIM, 6, rowk, colk, tid);

  // pq = Pinv q
  if (tid < NDIM) {
    float acc = 0.f;
    for (int k = 0; k < NDIM; ++k) acc += sP[tid * NDIM + k] * qv[k];
    pq[tid] = acc;
  }
  __syncthreads();

  // PinvHt(64x128) = Pinv * H^T  (4x8 tiles; uniform trip count)
#pragma unroll
  for (int i = 0; i < 4; ++i) {
    const int t = wave + i * 8;
    gemm_tile(sP, NDIM, 1, sH, 1, NDIM, sPHt, MDIM,
              (t >> 3) * 16, (t & 7) * 16, NDIM, lane);
  }
  __syncthreads();

  // hq = H pq - b
  if (tid < MDIM) {
    float acc = 0.f;
    for (int k = 0; k < NDIM; ++k) acc += sH[tid * NDIM + k] * pq[k];
    hq[tid] = acc - bv[tid];
  }
  __syncthreads();

  // S(128x128) = H * PinvHt ; then M = I + S
#pragma unroll
  for (int i = 0; i < 8; ++i) {
    const int t = wave + i * 8;
    gemm_tile(sH, NDIM, 1, sPHt, MDIM, 1, sTD, MDIM,
              (t >> 3) * 16, (t & 7) * 16, NDIM, lane);
  }
  __syncthreads();
  if (tid < MDIM) sTD[tid * MDIM + tid] += 1.0f;
  __syncthreads();

  // tD = (I + S)^{-1}
  sweep_inverse(sTD, MDIM, 7, rowk, colk, tid);

  // mu = tD hq
  if (tid < MDIM) {
    float acc = 0.f;
    for (int k = 0; k < MDIM; ++k) acc += sTD[tid * MDIM + k] * hq[k];
    mu[tid] = acc;
  }
  __syncthreads();

  // G(64x64) = H^T H
#pragma unroll
  for (int i = 0; i < 2; ++i) {
    const int t = wave + i * 8;
    gemm_tile(sH, 1, NDIM, sH, NDIM, 1, sP, NDIM,
              (t >> 2) * 16, (t & 3) * 16, MDIM, lane);
  }
  __syncthreads();
  sweep_inverse(sP, NDIM, 6, rowk, colk, tid);

  // pinvH(64x128) = Ginv * H^T  (overwrites PinvHt, no longer needed)
#pragma unroll
  for (int i = 0; i < 4; ++i) {
    const int t = wave + i * 8;
    gemm_tile(sP, NDIM, 1, sH, 1, NDIM, sPHt, MDIM,
              (t >> 3) * 16, (t & 7) * 16, NDIM, lane);
  }
  __syncthreads();

  // hb = pinvH b ; x = 0
  if (tid < NDIM) {
    float acc = 0.f;
    for (int k = 0; k < MDIM; ++k) acc += sPHt[tid * MDIM + k] * bv[k];
    hb[tid] = acc;
  }
  if (tid < MDIM) { x1[tid] = 0.f; x2[tid] = 0.f; }
  __syncthreads();

  // ---- preload loop-invariant A-operands into VGPRs ----
  // u-matvec: wave w owns rows [16w,16w+16) of tD : 32 k-steps (64 VGPRs).
  v2f aTD[32];
#pragma unroll
  for (int kk = 0; kk < 32; ++kk)
    aTD[kk] = *(const v2f*)&sTD[(wave * 16 + mr) * MDIM + 4 * kk + 2 * h];
  // p-matvec: wave w owns rows [16*(w&3),+16), K-half (w>>2) of pinvH (32 VGPRs).
  const int m0p = (wave & 3) * 16;
  const int kbp = (wave >> 2) * 64;
  v2f aPH[16];
#pragma unroll
  for (int kk = 0; kk < 16; ++kk)
    aPH[kk] = *(const v2f*)&sPHt[(m0p + mr) * MDIM + kbp + 4 * kk + 2 * h];

  const size_t xbase = (size_t)bid * (iters + 1) * TWOM;
  const size_t pbase = (size_t)bs * (iters + 1) * TWOM +
                       (size_t)bid * (iters + 1) * NDIM;
  for (int e = tid; e < TWOM; e += BLK) out[xbase + e] = 0.f;   // X0 = 0
  if (tid < NDIM) out[pbase + tid] = -hb[tid];                  // p0 = -pinvH b

  float* pv = rowk;  // 128 floats: two 64-float K-partials of p

  // ---- fixed-point iterations: WMMA from registers, vectors in LDS ----
  for (int t = 1; t <= iters; ++t) {
    __syncthreads();
    if (tid < MDIM) sv[tid] = x1[tid] + x2[tid];
    __syncthreads();

    // u = tD * s : wave w -> rows [16w,16w+16); s broadcast over B columns
    {
      v8f c = {};
#pragma unroll
      for (int kk = 0; kk < 32; ++kk) {
        v2f bb;
        bb.x = sv[4 * kk + 2 * h];
        bb.y = sv[4 * kk + 2 * h + 1];
        c = wmma4(aTD[kk], bb, c);
      }
      if (mr == 0) {
#pragma unroll
        for (int r = 0; r < 8; ++r) uv[wave * 16 + r + 8 * h] = c[r];
      }
    }
    __syncthreads();

    if (tid < MDIM) {
      float x1n = uv[tid] + mu[tid];
      float x2n = sv[tid] - 2.0f * x1n;
      x2n = x2n > 0.f ? x2n : 0.f;
      x1[tid] = x1n;
      x2[tid] = x2n;
      out[xbase + (size_t)t * TWOM + tid]        = x1n;
      out[xbase + (size_t)t * TWOM + MDIM + tid] = x2n;
    }
    __syncthreads();

    // p-partials: all 8 waves (4 row-tiles x 2 K-halves) of pinvH * x2
    {
      v8f c = {};
#pragma unroll
      for (int kk = 0; kk < 16; ++kk) {
        v2f bb;
        bb.x = x2[kbp + 4 * kk + 2 * h];
        bb.y = x2[kbp + 4 * kk + 2 * h + 1];
        c = wmma4(aPH[kk], bb, c);
      }
      if (mr == 0) {
#pragma unroll
        for (int r = 0; r < 8; ++r)
          pv[(wave >> 2) * NDIM + m0p + r + 8 * h] = c[r];
      }
    }
    __syncthreads();

    if (tid < NDIM)  // combine K-partials, coalesced store
      out[pbase + (size_t)t * NDIM + tid] = pv[tid] + pv[NDIM + tid] - hb[tid];
  }
}

extern "C" void kernel_launch(void* const* d_in, const int* in_sizes, int n_in,
                              void* d_out, int out_size, void* d_ws, size_t ws_size,
                              hipStream_t stream) {
  (void)n_in; (void)out_size; (void)d_ws; (void)ws_size;
  const float* q = (const float*)d_in[0];
  const float* b = (const float*)d_in[1];
  const float* P = (const float*)d_in[2];
  const float* H = (const float*)d_in[3];
  const int* iters = (const int*)d_in[4];
  float* out = (float*)d_out;

  const int bs = in_sizes[0] / NDIM;  // q is (bs, 64)

  const size_t shmem =
      (size_t)(MDIM * MDIM + MDIM * NDIM + NDIM * MDIM + NDIM * NDIM + 1344) *
      sizeof(float);  // ~153 KB, within 320 KB/WGP
  (void)hipFuncSetAttribute((const void*)QPSolver_kernel,
                            hipFuncAttributeMaxDynamicSharedMemorySize,
                            (int)shmem);

  QPSolver_kernel<<<bs, BLK, shmem, stream>>>(q, b, P, H, iters, out, bs);
}